// PointFeatureConv_62801011802167
// MI455X (gfx1250) — compile-verified
//
#include <hip/hip_runtime.h>
#include <hip/hip_bf16.h>
#include <math.h>

// ---------------------------------------------------------------------------
// PointFeatureConv on MI455X (gfx1250, wave32).
//   B=2, N=8192, C_IN=32, HID=128, C_OUT=64, KNN_K=16
// Restructured:
//   Fn = feat @ W1[0:32]          (WMMA f32 GEMM)
//   Fs = feat @ W1[32:64] + b1    (WMMA f32 GEMM)
//   per edge: h = Fn[j] + Fs[i] + rel_pos . W1[64:67]; LN; GELU; mean over K
//   out1 = Gmean @ W2 + b2        (WMMA f32 GEMM)   [mean commuted past W2]
//   h2   = out1 @ Wo1 + bo1       (WMMA f32 GEMM) ; LN ; GELU
//   out  = g2 @ Wo2 + bo2         (WMMA f32 GEMM)
// ---------------------------------------------------------------------------

#define BATCH   2
#define NPTS    8192
#define CIN     32
#define HID     128
#define COUT    64
#define KNN_K   16
#define BN      (BATCH * NPTS)
#define LN_EPS  1e-5f

typedef float v2f __attribute__((ext_vector_type(2)));
typedef float v8f __attribute__((ext_vector_type(8)));

#if defined(__HIP_DEVICE_COMPILE__) && \
    __has_builtin(__builtin_amdgcn_global_load_async_to_lds_b32) && \
    __has_builtin(__builtin_amdgcn_s_wait_asynccnt)
#define HAS_ASYNC_LDS 1
typedef __attribute__((address_space(1))) int*  gas1_t;   // builtin expects AS1 int*
typedef __attribute__((address_space(3))) int*  las3_t;   // builtin expects AS3 int*
#else
#define HAS_ASYNC_LDS 0
#endif

// ---------------------------------------------------------------------------
// Generic fp32 WMMA GEMM:  C[M,Nd] = A[M,Kd] @ W[Kd,Nd] (+ bias[Nd])
// One wave per 16x16 output tile; Kd stepped by 4 (V_WMMA_F32_16X16X4_F32).
// VGPR layouts per CDNA5 ISA 7.12.2:
//   A 16x4 : lane&15 -> M row; lanes>=16 hold K+2,K+3 (v2f = consecutive K)
//   B 4x16 : lane&15 -> N col; lanes>=16 hold K-rows 2,3
//   C 16x16: acc[v] holds M = v + (lane>>4)*8, N = lane&15
// Pointer-bump addressing so the 24-bit IOFFSET absorbs the k-walk.
// ---------------------------------------------------------------------------
__global__ void gemm_bias_wmma_f32(const float* __restrict__ A,
                                   const float* __restrict__ W,
                                   const float* __restrict__ bias,
                                   float* __restrict__ C,
                                   int M, int Kd, int Nd)
{
    const int wave = (blockIdx.x * blockDim.x + threadIdx.x) >> 5;
    const int lane = threadIdx.x & 31;
    const int ntx  = Nd >> 4;
    if (wave >= (M >> 4) * ntx) return;
    const int mt = wave / ntx;
    const int nt = wave - mt * ntx;

    const int mrow  = mt * 16 + (lane & 15);
    const int ncol  = nt * 16 + (lane & 15);
    const int khalf = (lane >> 4) * 2;     // upper half-wave carries K+2,K+3

    v8f acc = {};
    const float* aptr = A + (size_t)mrow * Kd + khalf;
    const float* wptr = W + (size_t)khalf * Nd + ncol;
    for (int k0 = 0; k0 < Kd; k0 += 4) {
        v2f a = *(const v2f*)aptr;           // 8B-aligned pair of K values
        v2f b;
        b[0] = wptr[0];
        b[1] = wptr[Nd];
        acc = __builtin_amdgcn_wmma_f32_16x16x4_f32(
                  false, a, false, b, (short)0, acc, false, false);
        aptr += 4;
        wptr += (size_t)4 * Nd;
    }

    const float bv = bias ? bias[ncol] : 0.0f;
    float* cptr = C + (size_t)(mt * 16 + (lane >> 4) * 8) * Nd + ncol;
#pragma unroll
    for (int v = 0; v < 8; ++v)
        cptr[(size_t)v * Nd] = acc[v] + bv;
}

// ---------------------------------------------------------------------------
// Brute-force KNN, one thread per query point, LDS-tiled candidate vertices
// (staged with GLOBAL_LOAD_ASYNC_TO_LDS when available), register
// insertion-sort top-16 (fully unrolled -> stays in VGPRs).
// ---------------------------------------------------------------------------
__global__ void knn_kernel(const float* __restrict__ verts,
                           int* __restrict__ idx_out)
{
    __shared__ float sx[256], sy[256], sz[256];
    const int gid = blockIdx.x * blockDim.x + threadIdx.x;   // [0, BN)
    const int b   = gid / NPTS;
    const int i   = gid - b * NPTS;
    const float* vb = verts + (size_t)b * NPTS * 3;

    const float qx = vb[i * 3 + 0];
    const float qy = vb[i * 3 + 1];
    const float qz = vb[i * 3 + 2];

    float bd[KNN_K];
    int   bi[KNN_K];
#pragma unroll
    for (int s = 0; s < KNN_K; ++s) { bd[s] = 3.4e38f; bi[s] = 0; }

    for (int tile = 0; tile < NPTS; tile += 256) {
        const int j = tile + threadIdx.x;
#if HAS_ASYNC_LDS
        // Async DMA global->LDS (tracked by ASYNCcnt, s_wait_asynccnt 0
        // before the barrier) -- keeps staging off the VALU/VGPR path.
        __builtin_amdgcn_global_load_async_to_lds_b32(
            (gas1_t)(vb + j * 3 + 0), (las3_t)&sx[threadIdx.x], 0, 0);
        __builtin_amdgcn_global_load_async_to_lds_b32(
            (gas1_t)(vb + j * 3 + 1), (las3_t)&sy[threadIdx.x], 0, 0);
        __builtin_amdgcn_global_load_async_to_lds_b32(
            (gas1_t)(vb + j * 3 + 2), (las3_t)&sz[threadIdx.x], 0, 0);
        __builtin_amdgcn_s_wait_asynccnt(0);
#else
        sx[threadIdx.x] = vb[j * 3 + 0];
        sy[threadIdx.x] = vb[j * 3 + 1];
        sz[threadIdx.x] = vb[j * 3 + 2];
#endif
        __syncthreads();
        for (int t = 0; t < 256; ++t) {
            const float dx = sx[t] - qx;
            const float dy = sy[t] - qy;
            const float dz = sz[t] - qz;
            const float d  = dx * dx + dy * dy + dz * dz;
            if (d < bd[KNN_K - 1]) {
                bd[KNN_K - 1] = d;
                bi[KNN_K - 1] = tile + t;
#pragma unroll
                for (int s = KNN_K - 1; s > 0; --s) {
                    if (bd[s] < bd[s - 1]) {
                        float td = bd[s]; bd[s] = bd[s - 1]; bd[s - 1] = td;
                        int   ti = bi[s]; bi[s] = bi[s - 1]; bi[s - 1] = ti;
                    }
                }
            }
        }
        __syncthreads();
    }
#pragma unroll
    for (int s = 0; s < KNN_K; ++s)
        idx_out[(size_t)gid * KNN_K + s] = bi[s];
}

// ---------------------------------------------------------------------------
// Wave32 helpers
// ---------------------------------------------------------------------------
__device__ __forceinline__ float wave_sum32(float v)
{
#pragma unroll
    for (int m = 16; m >= 1; m >>= 1) v += __shfl_xor(v, m, 32);
    return v;
}

__device__ __forceinline__ float gelu_tanh(float x)
{   // jax.nn.gelu default (approximate=True)
    const float x3 = x * x * x;
    return 0.5f * x * (1.0f + tanhf(0.7978845608f * (x + 0.044715f * x3)));
}

// ---------------------------------------------------------------------------
// Edge aggregation: one wave per point; lane owns 4 contiguous channels.
// Neighbor indices are fetched with ONE lane-parallel load and broadcast via
// __shfl; Fn rows are prefetched (global_prefetch_b8) before the k-loop.
// ---------------------------------------------------------------------------
__global__ void edge_agg_kernel(const float* __restrict__ Fn,
                                const float* __restrict__ Fs,
                                const int*   __restrict__ idx,
                                const float* __restrict__ verts,
                                const float* __restrict__ W1,   // [67,128]
                                const float* __restrict__ g1,
                                const float* __restrict__ be1,
                                float* __restrict__ Gmean)
{
    const int p = (blockIdx.x * blockDim.x + threadIdx.x) >> 5;
    if (p >= BN) return;
    const int lane = threadIdx.x & 31;
    const int c    = lane * 4;
    const int b    = p / NPTS;
    const int i    = p - b * NPTS;
    const float* vb = verts + (size_t)b * NPTS * 3;
    const float* Fnb = Fn + (size_t)b * NPTS * HID;

    // one coalesced load of all 16 neighbor ids (lanes 0..15), then shuffle
    const int jlane = idx[(size_t)p * KNN_K + (lane & (KNN_K - 1))];
    // prefetch the head of each neighbor's Fn row while we do setup math
    if (lane < KNN_K)
        __builtin_prefetch(Fnb + (size_t)jlane * HID, 0, 0);

    const float4 fs = *(const float4*)(Fs + (size_t)p * HID + c);
    const float4 wA = *(const float4*)(W1 + 64 * HID + c);
    const float4 wB = *(const float4*)(W1 + 65 * HID + c);
    const float4 wC = *(const float4*)(W1 + 66 * HID + c);
    const float4 gg = *(const float4*)(g1 + c);
    const float4 bb = *(const float4*)(be1 + c);

    const float qx = vb[i * 3 + 0];
    const float qy = vb[i * 3 + 1];
    const float qz = vb[i * 3 + 2];

    float ax = 0.f, ay = 0.f, az = 0.f, aw = 0.f;

#pragma unroll
    for (int k = 0; k < KNN_K; ++k) {
        const int j = __shfl(jlane, k, 32);
        const float4 fn = *(const float4*)(Fnb + (size_t)j * HID + c);
        const float rx = vb[j * 3 + 0] - qx;
        const float ry = vb[j * 3 + 1] - qy;
        const float rz = vb[j * 3 + 2] - qz;

        float hx = fn.x + fs.x + rx * wA.x + ry * wB.x + rz * wC.x;
        float hy = fn.y + fs.y + rx * wA.y + ry * wB.y + rz * wC.y;
        float hz = fn.z + fs.z + rx * wA.z + ry * wB.z + rz * wC.z;
        float hw = fn.w + fs.w + rx * wA.w + ry * wB.w + rz * wC.w;

        const float mu = wave_sum32(hx + hy + hz + hw) * (1.0f / HID);
        const float cx = hx - mu, cy = hy - mu, cz = hz - mu, cw = hw - mu;
        const float var = wave_sum32(cx * cx + cy * cy + cz * cz + cw * cw) * (1.0f / HID);
        const float rs = rsqrtf(var + LN_EPS);

        ax += gelu_tanh(cx * rs * gg.x + bb.x);
        ay += gelu_tanh(cy * rs * gg.y + bb.y);
        az += gelu_tanh(cz * rs * gg.z + bb.z);
        aw += gelu_tanh(cw * rs * gg.w + bb.w);
    }

    float4 o;
    o.x = ax * (1.0f / KNN_K);
    o.y = ay * (1.0f / KNN_K);
    o.z = az * (1.0f / KNN_K);
    o.w = aw * (1.0f / KNN_K);
    *(float4*)(Gmean + (size_t)p * HID + c) = o;
}

// ---------------------------------------------------------------------------
// LayerNorm + GELU over HID=128 channels, one wave per point.
// ---------------------------------------------------------------------------
__global__ void ln_gelu_kernel(const float* __restrict__ X,
                               const float* __restrict__ g,
                               const float* __restrict__ be,
                               float* __restrict__ Y)
{
    const int p = (blockIdx.x * blockDim.x + threadIdx.x) >> 5;
    if (p >= BN) return;
    const int lane = threadIdx.x & 31;
    const int c    = lane * 4;

    const float4 x  = *(const float4*)(X + (size_t)p * HID + c);
    const float4 gg = *(const float4*)(g + c);
    const float4 bb = *(const float4*)(be + c);

    const float mu = wave_sum32(x.x + x.y + x.z + x.w) * (1.0f / HID);
    const float cx = x.x - mu, cy = x.y - mu, cz = x.z - mu, cw = x.w - mu;
    const float var = wave_sum32(cx * cx + cy * cy + cz * cz + cw * cw) * (1.0f / HID);
    const float rs = rsqrtf(var + LN_EPS);

    float4 o;
    o.x = gelu_tanh(cx * rs * gg.x + bb.x);
    o.y = gelu_tanh(cy * rs * gg.y + bb.y);
    o.z = gelu_tanh(cz * rs * gg.z + bb.z);
    o.w = gelu_tanh(cw * rs * gg.w + bb.w);
    *(float4*)(Y + (size_t)p * HID + c) = o;
}

// ---------------------------------------------------------------------------
// Host-side orchestration
// ---------------------------------------------------------------------------
extern "C" void kernel_launch(void* const* d_in, const int* in_sizes, int n_in,
                              void* d_out, int out_size, void* d_ws, size_t ws_size,
                              hipStream_t stream)
{
    const float* vertices = (const float*)d_in[0];
    const float* features = (const float*)d_in[1];
    const float* W1  = (const float*)d_in[2];
    const float* b1  = (const float*)d_in[3];
    const float* g1  = (const float*)d_in[4];
    const float* be1 = (const float*)d_in[5];
    const float* W2  = (const float*)d_in[6];
    const float* b2  = (const float*)d_in[7];
    const float* Wo1 = (const float*)d_in[8];
    const float* bo1 = (const float*)d_in[9];
    const float* go  = (const float*)d_in[10];
    const float* beo = (const float*)d_in[11];
    const float* Wo2 = (const float*)d_in[12];
    const float* bo2 = (const float*)d_in[13];
    float* out = (float*)d_out;

    // Workspace carve-up (bytes):
    //  [0,1MB)    idx         BN*16 int
    //  [1,9MB)    Fn          BN*128 f32   (reused as h2raw after edge pass)
    //  [9,17MB)   Fs          BN*128 f32   (reused as g2 after edge pass)
    //  [17,25MB)  Gmean       BN*128 f32
    //  [25,29MB)  out1        BN*64  f32
    char* ws = (char*)d_ws;
    int*   idxw = (int*)ws;
    float* Fn   = (float*)(ws + (size_t)1  * (1 << 20));
    float* Fs   = (float*)(ws + (size_t)9  * (1 << 20));
    float* Gm   = (float*)(ws + (size_t)17 * (1 << 20));
    float* o1   = (float*)(ws + (size_t)25 * (1 << 20));
    float* h2   = Fn;   // reuse: Fn dead after edge_agg
    float* g2   = Fs;   // reuse: Fs dead after edge_agg

    const int THREADS = 256;                 // 8 wave32 waves / block
    const int WPB = THREADS / 32;

    // 1) KNN: idx[BN,16]
    knn_kernel<<<BN / THREADS, THREADS, 0, stream>>>(vertices, idxw);

    // 2) Fn = feat @ W1[0:32]         (M=BN, K=32, N=128)
    {
        int tiles = (BN / 16) * (HID / 16);
        gemm_bias_wmma_f32<<<(tiles + WPB - 1) / WPB, THREADS, 0, stream>>>(
            features, W1, nullptr, Fn, BN, CIN, HID);
    }
    // 3) Fs = feat @ W1[32:64] + b1
    {
        int tiles = (BN / 16) * (HID / 16);
        gemm_bias_wmma_f32<<<(tiles + WPB - 1) / WPB, THREADS, 0, stream>>>(
            features, W1 + (size_t)CIN * HID, b1, Fs, BN, CIN, HID);
    }
    // 4) per-edge LN+GELU + mean over K  -> Gmean[BN,128]
    edge_agg_kernel<<<(BN * 32) / THREADS, THREADS, 0, stream>>>(
        Fn, Fs, idxw, vertices, W1, g1, be1, Gm);

    // 5) out1 = Gmean @ W2 + b2       (M=BN, K=128, N=64)
    {
        int tiles = (BN / 16) * (COUT / 16);
        gemm_bias_wmma_f32<<<(tiles + WPB - 1) / WPB, THREADS, 0, stream>>>(
            Gm, W2, b2, o1, BN, HID, COUT);
    }
    // 6) h2 = out1 @ Wo1 + bo1        (M=BN, K=64, N=128)
    {
        int tiles = (BN / 16) * (HID / 16);
        gemm_bias_wmma_f32<<<(tiles + WPB - 1) / WPB, THREADS, 0, stream>>>(
            o1, Wo1, bo1, h2, BN, COUT, HID);
    }
    // 7) g2 = GELU(LN(h2; go, beo))
    ln_gelu_kernel<<<(BN * 32) / THREADS, THREADS, 0, stream>>>(h2, go, beo, g2);

    // 8) out = g2 @ Wo2 + bo2         (M=BN, K=128, N=64) -> d_out
    {
        int tiles = (BN / 16) * (COUT / 16);
        gemm_bias_wmma_f32<<<(tiles + WPB - 1) / WPB, THREADS, 0, stream>>>(
            g2, Wo2, bo2, out, BN, HID, COUT);
    }
}